// Quantizer_781684048560
// MI455X (gfx1250) — compile-verified
//
#include <hip/hip_runtime.h>
#include <hip/hip_bf16.h>
#include <math.h>

typedef __attribute__((ext_vector_type(2))) float v2f;
typedef __attribute__((ext_vector_type(8))) float v8f;

#define N_E        4096
#define EDIM       512
#define NROW       16384        // 32 * 512 latent vectors
#define ROWS_PER_WG 16
#define ZT_STRIDE  (EDIM + 4)   // 516: rotates LDS banks by 4 per row -> conflict-free
#define WAVES      8

// ---------------------------------------------------------------------------
// ws layout (floats): [0, N_E)   = counts histogram
//                     [N_E]      = loss accumulator (sum of squared diffs)
//                     [N_E+1, +N_E) = ||emb||^2 per code
// ---------------------------------------------------------------------------

__global__ void vq_init(float* ws) {
    int i = blockIdx.x * blockDim.x + threadIdx.x;
    if (i <= N_E) ws[i] = 0.0f;   // counts + loss accumulator
}

__global__ void vq_enorm(const float* __restrict__ emb, float* __restrict__ enorm) {
    int c = blockIdx.x * blockDim.x + threadIdx.x;
    if (c < N_E) {
        const float4* row = (const float4*)(emb + (size_t)c * EDIM);
        float s = 0.0f;
        #pragma unroll 4
        for (int i = 0; i < EDIM / 4; ++i) {
            float4 v = row[i];
            s += v.x * v.x + v.y * v.y + v.z * v.z + v.w * v.w;
        }
        enorm[c] = s;
    }
}

__launch_bounds__(256)
__global__ void vq_main(const float* __restrict__ z,
                        const float* __restrict__ emb,
                        const float* __restrict__ enorm,
                        float* __restrict__ counts,
                        float* __restrict__ loss_acc,
                        float* __restrict__ out_zq,
                        float* __restrict__ out_idx) {
    __shared__ float zt[ROWS_PER_WG * ZT_STRIDE];          // 33,024 B
    __shared__ float redv[WAVES * 16 * 16];                //  8,192 B
    __shared__ int   redi[WAVES * 16 * 16];                //  8,192 B
    __shared__ int   best_sh[ROWS_PER_WG];
    __shared__ float wsum[WAVES];

    const int tid  = threadIdx.x;
    const int wave = tid >> 5;
    const int lane = tid & 31;
    const int r    = lane & 15;     // row index within 16-lane half
    const int hi   = lane >> 4;     // half select (K pair / M+8)
    const int m0   = blockIdx.x * ROWS_PER_WG;

    // ---- stage 16 z rows into LDS (float4 cooperative load) ----
    for (int e = tid; e < ROWS_PER_WG * (EDIM / 4); e += 256) {
        int row = e / (EDIM / 4);
        int c4  = e % (EDIM / 4);
        float4 v = ((const float4*)(z + (size_t)(m0 + row) * EDIM))[c4];
        float* dst = &zt[row * ZT_STRIDE + c4 * 4];
        dst[0] = v.x; dst[1] = v.y; dst[2] = v.z; dst[3] = v.w;
    }
    __syncthreads();

    // Per-lane running argmin: VGPR j of the accumulator holds D[M][N] with
    // M = j + 8*hi, N = r  (32-bit C/D 16x16 layout from the ISA).
    float minv[8];
    int   mini[8];
    #pragma unroll
    for (int j = 0; j < 8; ++j) { minv[j] = 3.4e38f; mini[j] = 0x7fffffff; }

    // A-matrix 16x4 fp32 layout: lane (r,hi) holds A[r][2*hi], A[r][2*hi+1]
    const float* zrow = &zt[r * ZT_STRIDE + 2 * hi];

    for (int chunk = wave; chunk < N_E / 16; chunk += WAVES) {
        const int n0 = chunk * 16;
        // B-matrix 4x16 fp32 layout: lane (r,hi) holds B[2*hi][r], B[2*hi+1][r]
        // = emb[n0+r][k+2*hi], emb[n0+r][k+2*hi+1]   (B = emb^T)
        const float* erow = emb + (size_t)(n0 + r) * EDIM + 2 * hi;

        v8f acc = {};
        #pragma unroll 8
        for (int k = 0; k < EDIM; k += 4) {
            v2f a = *(const v2f*)(zrow + k);   // ds_load_b64, 8B aligned
            v2f b = *(const v2f*)(erow + k);   // global_load_b64, 8B aligned
            acc = __builtin_amdgcn_wmma_f32_16x16x4_f32(
                false, a, false, b, (short)0, acc, false, false);
        }

        const float en   = enorm[n0 + r];      // this lane's column is code n0+r
        const int   code = n0 + r;
        #pragma unroll
        for (int j = 0; j < 8; ++j) {
            float dist = en - 2.0f * acc[j];   // ||z||^2 constant per row: dropped
            if (dist < minv[j]) { minv[j] = dist; mini[j] = code; }
        }
    }

    // ---- cross-lane / cross-wave argmin reduction via LDS ----
    #pragma unroll
    for (int j = 0; j < 8; ++j) {
        int row = j + 8 * hi;
        int o   = (wave * 16 + row) * 16 + r;
        redv[o] = minv[j];
        redi[o] = mini[j];
    }
    __syncthreads();

    if (tid < ROWS_PER_WG) {
        float bv = 3.4e38f;
        int   bi = 0x7fffffff;
        for (int w = 0; w < WAVES; ++w) {
            for (int c = 0; c < 16; ++c) {
                int   o = (w * 16 + tid) * 16 + c;
                float v = redv[o];
                int   i = redi[o];
                if (v < bv || (v == bv && i < bi)) { bv = v; bi = i; }  // first-occurrence tie-break
            }
        }
        best_sh[tid] = bi;
        out_idx[m0 + tid] = (float)bi;
        atomicAdd(&counts[bi], 1.0f);
    }
    __syncthreads();

    // ---- gather z_q = emb[idx], emit, and accumulate loss sum ----
    float lsum = 0.0f;
    for (int e = tid; e < ROWS_PER_WG * EDIM; e += 256) {
        int row = e >> 9;            // EDIM == 512
        int col = e & (EDIM - 1);
        float q = emb[(size_t)best_sh[row] * EDIM + col];
        float d = q - zt[row * ZT_STRIDE + col];
        lsum += d * d;
        out_zq[(size_t)(m0 + row) * EDIM + col] = q;  // z_q_st == z_q numerically
    }
    #pragma unroll
    for (int off = 16; off > 0; off >>= 1)
        lsum += __shfl_down(lsum, off, 32);
    if (lane == 0) wsum[wave] = lsum;
    __syncthreads();
    if (tid == 0) {
        float s = 0.0f;
        #pragma unroll
        for (int w = 0; w < WAVES; ++w) s += wsum[w];
        atomicAdd(loss_acc, s);
    }
}

__global__ void vq_final(const float* __restrict__ counts,
                         const float* __restrict__ loss_acc,
                         float* __restrict__ out_loss,
                         float* __restrict__ out_perp) {
    __shared__ float sred[256];
    int tid = threadIdx.x;
    float s = 0.0f;
    for (int i = tid; i < N_E; i += 256) {
        float p = counts[i] * (1.0f / (float)NROW);
        s += p * logf(p + 1e-10f);
    }
    sred[tid] = s;
    __syncthreads();
    for (int off = 128; off > 0; off >>= 1) {
        if (tid < off) sred[tid] += sred[tid + off];
        __syncthreads();
    }
    if (tid == 0) {
        *out_perp = expf(-sred[0]);
        // loss = (1 + BETA) * mean((z_q - z)^2); stop_gradient is a no-op on values
        *out_loss = loss_acc[0] * (1.25f / ((float)NROW * (float)EDIM));
    }
}

extern "C" void kernel_launch(void* const* d_in, const int* in_sizes, int n_in,
                              void* d_out, int out_size, void* d_ws, size_t ws_size,
                              hipStream_t stream) {
    (void)in_sizes; (void)n_in; (void)out_size; (void)ws_size;
    const float* z   = (const float*)d_in[0];   // (32, 512, 512) f32
    const float* emb = (const float*)d_in[1];   // (4096, 512) f32

    float* out      = (float*)d_out;
    float* out_loss = out;                                   // [0]
    float* out_zq   = out + 1;                               // [1 .. 1+NROW*EDIM)
    float* out_idx  = out + 1 + (size_t)NROW * EDIM;         // indices as float
    float* out_perp = out + 1 + (size_t)NROW * EDIM + NROW;  // last element

    float* ws       = (float*)d_ws;
    float* counts   = ws;                // N_E floats
    float* loss_acc = ws + N_E;          // 1 float
    float* enorm    = ws + N_E + 1;      // N_E floats

    vq_init <<<(N_E + 1 + 255) / 256, 256, 0, stream>>>(ws);
    vq_enorm<<<N_E / 256,            256, 0, stream>>>(emb, enorm);
    vq_main <<<NROW / ROWS_PER_WG,   256, 0, stream>>>(z, emb, enorm, counts,
                                                       loss_acc, out_zq, out_idx);
    vq_final<<<1,                    256, 0, stream>>>(counts, loss_acc,
                                                       out_loss, out_perp);
}